// GraphVAE_52175262712009
// MI455X (gfx1250) — compile-verified
//
#include <hip/hip_runtime.h>
#include <hip/hip_bf16.h>
#include <math.h>

#define NN 96
#define D_IN 16
#define HID 256
#define LAT 128
#define EDGE 4656
#define OUT_DIM 6192
#define XP 100   // pitched LDS row (100 dwords -> conflict-free for 16-lane column reads)

typedef __attribute__((ext_vector_type(2))) float v2f;
typedef __attribute__((ext_vector_type(8))) float v8f;

// ---------------- generic f32 WMMA GEMM: C(MxN) = A(MxK) @ B(KxN), row-major ----
// one 16x16 C tile per wave; K stepped by 4 via V_WMMA_F32_16X16X4_F32
__global__ void gemm_f32_wmma(const float* __restrict__ A, const float* __restrict__ B,
                              float* __restrict__ C, int M, int K, int Nd) {
  int wid  = (blockIdx.x * blockDim.x + threadIdx.x) >> 5;
  int lane = threadIdx.x & 31;
  int nt = Nd >> 4, mt = M >> 4;
  if (wid >= mt * nt) return;            // wave-uniform exit (EXEC stays all-ones for WMMA)
  int m0 = (wid / nt) * 16, n0 = (wid % nt) * 16;
  int half = lane >> 4, lm = lane & 15;
  v8f acc = {0.f,0.f,0.f,0.f,0.f,0.f,0.f,0.f};
  for (int k = 0; k < K; k += 4) {
    v2f a, b;
    // A frag 16x4: lanes 0-15 hold K={k,k+1}, lanes 16-31 hold K={k+2,k+3}, M=lane&15
    a.x = A[(m0 + lm) * K + k + 2 * half];
    a.y = A[(m0 + lm) * K + k + 2 * half + 1];
    // B frag 4x16: row striped across lanes within a VGPR, K split across lane halves
    b.x = B[(k + 2 * half) * Nd + n0 + lm];
    b.y = B[(k + 2 * half + 1) * Nd + n0 + lm];
    acc = __builtin_amdgcn_wmma_f32_16x16x4_f32(false, a, false, b, (short)0, acc, false, false);
  }
  for (int v = 0; v < 8; ++v)            // C/D layout: row = v + 8*half, col = lane&15
    C[(m0 + v + 8 * half) * Nd + n0 + lm] = acc[v];
}

// ---------------- BatchNorm (training mode, over node axis) + ReLU --------------
__global__ void bn_relu(const float* __restrict__ h, const float* __restrict__ g,
                        const float* __restrict__ b, float* __restrict__ x) {
  int c = threadIdx.x;                   // 256 columns
  float s = 0.f;
  for (int r = 0; r < NN; ++r) s += h[r * HID + c];
  float mu = s * (1.f / NN);
  float v = 0.f;
  for (int r = 0; r < NN; ++r) { float d = h[r * HID + c] - mu; v += d * d; }
  v *= (1.f / NN);
  float is = 1.f / sqrtf(v + 1e-5f);
  float gc = g[c], bc = b[c];
  for (int r = 0; r < NN; ++r) {
    float y = (h[r * HID + c] - mu) * is * gc + bc;
    x[r * HID + c] = fmaxf(y, 0.f);
  }
}

// ---------------- BatchNorm + sum-pool -> gh[HID] -------------------------------
__global__ void bn_pool(const float* __restrict__ h, const float* __restrict__ g,
                        const float* __restrict__ b, float* __restrict__ gh) {
  int c = threadIdx.x;
  float s = 0.f;
  for (int r = 0; r < NN; ++r) s += h[r * HID + c];
  float mu = s * (1.f / NN);
  float v = 0.f;
  for (int r = 0; r < NN; ++r) { float d = h[r * HID + c] - mu; v += d * d; }
  v *= (1.f / NN);
  float is = 1.f / sqrtf(v + 1e-5f);
  float gc = g[c], bc = b[c], sum = 0.f;
  for (int r = 0; r < NN; ++r) sum += (h[r * HID + c] - mu) * is * gc + bc;
  gh[c] = sum;
}

// ---------------- VAE head: z_mu, z_ls, z, hr = relu(d1 z + b) ------------------
__global__ void vae_head(const float* __restrict__ gh,
                         const float* __restrict__ e11w, const float* __restrict__ e11b,
                         const float* __restrict__ e12w, const float* __restrict__ e12b,
                         const float* __restrict__ d1w, const float* __restrict__ d1b,
                         const float* __restrict__ eps,
                         float* __restrict__ zmu, float* __restrict__ zls,
                         float* __restrict__ hr) {
  __shared__ float zsh[LAT];
  int t = threadIdx.x;                   // 256 threads
  if (t < LAT) {
    float s = e11b[t];
    for (int j = 0; j < HID; ++j) s += e11w[t * HID + j] * gh[j];
    zmu[t] = s;
  } else {
    int i = t - LAT;
    float s = e12b[i];
    for (int j = 0; j < HID; ++j) s += e12w[i * HID + j] * gh[j];
    zls[i] = s;
  }
  __syncthreads();
  if (t < LAT) zsh[t] = eps[t] * expf(0.5f * zls[t]) + zmu[t];
  __syncthreads();
  if (t < LAT) {
    float s = d1b[t];
    for (int j = 0; j < LAT; ++j) s += d1w[t * LAT + j] * zsh[j];
    hr[t] = fmaxf(s, 0.f);
  }
}

// ---------------- decoder layer 2 + sigmoid on edge logits ----------------------
__global__ void decoder2(const float* __restrict__ d2w, const float* __restrict__ d2b,
                         const float* __restrict__ hr, float* __restrict__ hdec,
                         float* __restrict__ outp) {
  int o = blockIdx.x * blockDim.x + threadIdx.x;
  if (o >= OUT_DIM) return;
  float s = d2b[o];
  for (int j = 0; j < LAT; ++j) s += d2w[o * LAT + j] * hr[j];
  hdec[o] = s;
  if (o < EDGE) outp[o] = 1.f / (1.f + expf(-s));
}

// ---------------- recon / degrees / A,B,Sdiag factors ---------------------------
__global__ void prep(const float* __restrict__ adj, const float* __restrict__ outp,
                     float* __restrict__ recon, float* __restrict__ degt,
                     float* __restrict__ degr, float* __restrict__ Am,
                     float* __restrict__ Bm, float* __restrict__ Sd) {
  int t = threadIdx.x;                   // 512 threads, one block
  for (int e = t; e < NN * NN; e += blockDim.x) {
    int i = e / NN, j = e % NN;
    int lo = i < j ? i : j, hi = i < j ? j : i;
    int idx = lo * NN - (lo * (lo - 1)) / 2 + (hi - lo);   // row-major triu index
    recon[e] = outp[idx];
  }
  __syncthreads();
  if (t < NN) {
    float st = 0.f, sr = 0.f;
    for (int j = 0; j < NN; ++j) { st += adj[t * NN + j]; sr += recon[t * NN + j]; }
    degt[t] = st; degr[t] = sr;
  }
  __syncthreads();
  for (int e = t; e < NN * NN; e += blockDim.x) {
    int i = e / NN, j = e % NN;
    float off = (i == j) ? 0.f : 1.f;
    Am[e] = adj[e]   * adj[i * NN + i]   * adj[j * NN + j]   * off;
    Bm[e] = recon[e] * recon[i * NN + i] * recon[j * NN + j] * off;
    // Sdiag[i,a] = dt[i]*dr[a]*sim[i,a]   (here j plays the role of a)
    Sd[e] = adj[i * NN + i] * recon[j * NN + j] *
            (1.f / (fabsf(degt[i] - degr[j]) + 1.f));
  }
}

// ---------------- 50-iteration max-pooling matching, single block ---------------
// structured mpm: Mt[j,a] = max_b B[a,b]*x[j,b];  xn = x.*Sdiag + A @ Mt; normalize
__global__ void __launch_bounds__(1024)
mpm(const float* __restrict__ Am, const float* __restrict__ Bm,
    const float* __restrict__ Sd, float* __restrict__ Mt,
    float* __restrict__ xn, float* __restrict__ asn) {
  __shared__ float xs[NN * XP];
  __shared__ float sred;
  int t = threadIdx.x;
  int lane = t & 31, wid = t >> 5;
  int half = lane >> 4, lm = lane & 15;
  for (int e = t; e < NN * NN; e += blockDim.x)
    xs[(e / NN) * XP + (e % NN)] = 1.0f / NN;
  for (int it = 0; it < 50; ++it) {
    __syncthreads();
    if (t == 0) sred = 0.f;
    // step 1: Mt[j,a] (nonneg, includes b==a term which is 0 since B diag is 0)
    for (int e = t; e < NN * NN; e += blockDim.x) {
      int j = e / NN, a = e % NN;
      const float* br = Bm + a * NN;
      const float* xr = xs + j * XP;
      float m = 0.f;
      for (int b = 0; b < NN; ++b) m = fmaxf(m, br[b] * xr[b]);
      Mt[j * NN + a] = m;
    }
    __syncthreads();
    // step 2: xn = x.*Sdiag + A @ Mt via WMMA, 36 tiles over 32 waves
    float ss = 0.f;
    for (int tile = wid; tile < 36; tile += 32) {
      int m0 = (tile / 6) * 16, n0 = (tile % 6) * 16;
      v8f acc = {0.f,0.f,0.f,0.f,0.f,0.f,0.f,0.f};
      for (int k = 0; k < NN; k += 4) {
        v2f a, b;
        a.x = Am[(m0 + lm) * NN + k + 2 * half];
        a.y = Am[(m0 + lm) * NN + k + 2 * half + 1];
        b.x = Mt[(k + 2 * half) * NN + n0 + lm];
        b.y = Mt[(k + 2 * half + 1) * NN + n0 + lm];
        acc = __builtin_amdgcn_wmma_f32_16x16x4_f32(false, a, false, b, (short)0, acc,
                                                    false, false);
      }
      for (int v = 0; v < 8; ++v) {
        int row = m0 + v + 8 * half, col = n0 + lm;
        float xv = xs[row * XP + col] * Sd[row * NN + col] + acc[v];
        xn[row * NN + col] = xv;
        ss += xv * xv;
      }
    }
    atomicAdd(&sred, ss);
    __syncthreads();
    float rn = 1.f / sqrtf(sred);
    for (int e = t; e < NN * NN; e += blockDim.x)
      xs[(e / NN) * XP + (e % NN)] = xn[e] * rn;
  }
  __syncthreads();
  for (int e = t; e < NN * NN; e += blockDim.x)
    asn[e] = xs[(e / NN) * XP + (e % NN)];
}

// ---------------- greedy max-weight assignment (matches jnp.argmax ties) --------
__global__ void greedy(const float* __restrict__ asn, int* __restrict__ perm) {
  __shared__ float sc[NN * NN];
  __shared__ float rv[128];
  __shared__ int   ri[128];
  __shared__ int   rrc[2];
  const float NEG = -__builtin_inff();
  int t = threadIdx.x;                   // 128 threads
  for (int e = t; e < NN * NN; e += 128) sc[e] = asn[e];
  __syncthreads();
  for (int n = 0; n < NN; ++n) {
    float best = NEG; int bi = NN * NN;
    for (int e = t; e < NN * NN; e += 128) {
      float v = sc[e];
      if (v > best) { best = v; bi = e; }   // first occurrence within thread
    }
    rv[t] = best; ri[t] = bi;
    __syncthreads();
    if (t == 0) {
      float bb = rv[0]; int ii = ri[0];
      for (int k = 1; k < 128; ++k)
        if (rv[k] > bb || (rv[k] == bb && ri[k] < ii)) { bb = rv[k]; ii = ri[k]; }
      int r = ii / NN, c = ii % NN;
      perm[c] = r; rrc[0] = r; rrc[1] = c;
    }
    __syncthreads();
    int r = rrc[0], c = rrc[1];
    if (t < NN) { sc[r * NN + t] = NEG; sc[t * NN + c] = NEG; }
    __syncthreads();
  }
}

// ---------------- final fused loss: bce + kl + fea ------------------------------
__global__ void loss_kernel(const float* __restrict__ feats, const float* __restrict__ adj,
                            const float* __restrict__ hdec, const float* __restrict__ outp,
                            const float* __restrict__ zmu, const float* __restrict__ zls,
                            const int* __restrict__ perm, float* __restrict__ dout) {
  __shared__ float accf, accb, acck;
  __shared__ int rowstart[NN];
  __shared__ int ind[NN];
  int t = threadIdx.x;                   // 256 threads
  if (t == 0) { accf = 0.f; accb = 0.f; acck = 0.f; }
  if (t < NN) { ind[t] = perm[t]; rowstart[t] = t * NN - (t * (t - 1)) / 2; }
  __syncthreads();
  float lf = 0.f;
  for (int e = t; e < NN * D_IN; e += blockDim.x) {
    int n = e / D_IN, d = e % D_IN;
    float diff = hdec[EDGE + e] - feats[ind[n] * D_IN + d];
    lf += diff * diff;
  }
  float lb = 0.f;
  for (int e = t; e < EDGE; e += blockDim.x) {
    int i = 0;
    while (i + 1 < NN && rowstart[i + 1] <= e) ++i;
    int j = i + (e - rowstart[i]);
    float tt = adj[ind[i] * NN + ind[j]];
    float o = outp[e];
    float lg = fmaxf(logf(o),    -100.f);
    float l1 = fmaxf(log1pf(-o), -100.f);
    lb += -(tt * lg + (1.f - tt) * l1);
  }
  float lk = 0.f;
  if (t < LAT) {
    float m = zmu[t], l = zls[t];
    lk = 1.f + l - m * m - expf(l);
  }
  atomicAdd(&accf, lf);
  atomicAdd(&accb, lb);
  atomicAdd(&acck, lk);
  __syncthreads();
  if (t == 0)
    dout[0] = accb / (float)EDGE + (-0.5f * acck / (float)(NN * NN)) + accf / (float)(NN * D_IN);
}

extern "C" void kernel_launch(void* const* d_in, const int* in_sizes, int n_in,
                              void* d_out, int out_size, void* d_ws, size_t ws_size,
                              hipStream_t stream) {
  const float* feats = (const float*)d_in[0];
  const float* adj   = (const float*)d_in[1];
  const float* c1w   = (const float*)d_in[2];
  const float* c2w   = (const float*)d_in[3];
  const float* bn1g  = (const float*)d_in[4];
  const float* bn1b  = (const float*)d_in[5];
  const float* bn2g  = (const float*)d_in[6];
  const float* bn2b  = (const float*)d_in[7];
  const float* e11w  = (const float*)d_in[8];
  const float* e11b  = (const float*)d_in[9];
  const float* e12w  = (const float*)d_in[10];
  const float* e12b  = (const float*)d_in[11];
  const float* d1w   = (const float*)d_in[12];
  const float* d1b   = (const float*)d_in[13];
  const float* d2w   = (const float*)d_in[14];
  const float* d2b   = (const float*)d_in[15];
  const float* eps   = (const float*)d_in[16];

  float* W    = (float*)d_ws;
  float* t0   = W;            // 96x16
  float* h1   = W + 2048;     // 96x256
  float* x1   = W + 26624;    // 96x256
  float* t1   = W + 51200;    // 96x256
  float* h2   = W + 75776;    // 96x256
  float* gh   = W + 100352;   // 256
  float* zmu  = W + 100608;   // 128
  float* zls  = W + 100736;   // 128
  float* hr   = W + 100864;   // 128
  float* hdec = W + 101120;   // 6192
  float* outp = W + 107520;   // 4656
  float* recon= W + 112384;   // 96x96
  float* degt = W + 121600;   // 96
  float* degr = W + 121728;   // 96
  float* Amat = W + 121856;   // 96x96
  float* Bmat = W + 131072;   // 96x96
  float* Sd   = W + 140288;   // 96x96
  float* Mt   = W + 149504;   // 96x96
  float* xnw  = W + 158720;   // 96x96
  float* asn  = W + 167936;   // 96x96
  int*   perm = (int*)(W + 177152); // 96

  // GCN encoder (all f32 WMMA)
  gemm_f32_wmma<<<1, 256, 0, stream>>>(adj, feats, t0, NN, NN, D_IN);     // 6 tiles
  gemm_f32_wmma<<<12, 256, 0, stream>>>(t0, c1w, h1, NN, D_IN, HID);      // 96 tiles
  bn_relu<<<1, 256, 0, stream>>>(h1, bn1g, bn1b, x1);
  gemm_f32_wmma<<<12, 256, 0, stream>>>(adj, x1, t1, NN, NN, HID);        // 96 tiles
  gemm_f32_wmma<<<12, 256, 0, stream>>>(t1, c2w, h2, NN, HID, HID);       // 96 tiles
  bn_pool<<<1, 256, 0, stream>>>(h2, bn2g, bn2b, gh);
  // VAE
  vae_head<<<1, 256, 0, stream>>>(gh, e11w, e11b, e12w, e12b, d1w, d1b, eps, zmu, zls, hr);
  decoder2<<<25, 256, 0, stream>>>(d2w, d2b, hr, hdec, outp);
  // graph matching (structured, no N^4 tensor)
  prep<<<1, 512, 0, stream>>>(adj, outp, recon, degt, degr, Amat, Bmat, Sd);
  mpm<<<1, 1024, 0, stream>>>(Amat, Bmat, Sd, Mt, xnw, asn);
  greedy<<<1, 128, 0, stream>>>(asn, perm);
  loss_kernel<<<1, 256, 0, stream>>>(feats, adj, hdec, outp, zmu, zls, perm, (float*)d_out);
}